// VoxelGrid_6579889897707
// MI455X (gfx1250) — compile-verified
//
#include <hip/hip_runtime.h>

// ---------------------------------------------------------------------------
// VoxelGrid scatter-mean for MI455X (gfx1250, wave32).
// Memory-bound op: no GEMM structure -> no WMMA applies. Strategy:
//   (1) hipMemsetAsync d_out to 0 (graph-capturable; resets state per replay).
//   (2) scatter: native non-returning global_atomic_add_f32 of
//       [xyz, feats, 1] into the 10-channel output record (sums in ch0..5,
//       count in ch9 == the reference's "ones" channel). AoS layout keeps each
//       point's 7 atomics within 1-2 cache lines; the whole 160MB accumulator
//       fits MI455X's 192MB L2, so atomics and the finalize read stay on-chip.
//   (3) finalize in-place: mean = sum / max(cnt,1), index-grid, occupancy.
// Roofline: ~480MB total traffic -> ~20us at 23.3TB/s HBM, less with L2 hits.
// ---------------------------------------------------------------------------

namespace vg {
constexpr int   V  = 100;          // voxel_size
constexpr int   D  = 102;          // padded dim
constexpr int   B  = 4;
constexpr int   N  = 65536;
constexpr int   C  = 10;           // output channels
constexpr long long NPTS = (long long)B * N;          // 262,144
constexpr long long NVOX = (long long)B * V * V * V;  // 4,000,000

// Match the reference's constant folding: python doubles -> f32 at use site.
constexpr double RES_D   = 2.0 / (100.0 + 1e-12);     // (bb_max-bb_min)/(V+eps)
constexpr double SHIFT_D = -1.0 - RES_D;              // bb_min - res
constexpr double DEN_D   = RES_D + 1e-12;             // res + eps
} // namespace vg

__global__ void __launch_bounds__(256)
voxel_scatter(const float* __restrict__ coords,
              const float* __restrict__ feats,
              float* __restrict__ out)
{
    const int i = blockIdx.x * blockDim.x + threadIdx.x;
    if (i >= (int)vg::NPTS) return;

    const int e = 3 * i;

    // Bring the feature cacheline into near caches (WGP-scope
    // global_prefetch_b8) while the index math + bounds branch resolve.
    __builtin_prefetch(&feats[e], 0, 3);

    const int b = i >> 16;                       // N == 65536

    const float shift = (float)vg::SHIFT_D;
    const float den   = (float)vg::DEN_D;

    const float px = coords[e + 0];
    const float py = coords[e + 1];
    const float pz = coords[e + 2];

    // idx = clip(floor((p - (bb_min - res)) / (res + eps)), 0, D-1), then
    // shift to cropped frame; skip points that land in the cropped border.
    // Keep true f32 division to stay bit-faithful to the reference's floor().
    int ix = (int)floorf((px - shift) / den);
    int iy = (int)floorf((py - shift) / den);
    int iz = (int)floorf((pz - shift) / den);
    ix = min(max(ix, 0), vg::D - 1) - 1;
    iy = min(max(iy, 0), vg::D - 1) - 1;
    iz = min(max(iz, 0), vg::D - 1) - 1;
    if ((unsigned)ix >= (unsigned)vg::V ||
        (unsigned)iy >= (unsigned)vg::V ||
        (unsigned)iz >= (unsigned)vg::V) return;

    const float f0 = feats[e + 0];
    const float f1 = feats[e + 1];
    const float f2 = feats[e + 2];

    float* p = out + (size_t)(((b * vg::V + ix) * vg::V + iy) * vg::V + iz) * vg::C;

    // Non-returning float atomics -> global_atomic_add_f32 (STOREcnt).
    unsafeAtomicAdd(p + 0, px);
    unsafeAtomicAdd(p + 1, py);
    unsafeAtomicAdd(p + 2, pz);
    unsafeAtomicAdd(p + 3, f0);
    unsafeAtomicAdd(p + 4, f1);
    unsafeAtomicAdd(p + 5, f2);
    unsafeAtomicAdd(p + 9, 1.0f);   // ones channel == count
}

__global__ void __launch_bounds__(256)
voxel_finalize(float* __restrict__ out)
{
    const long long v = (long long)blockIdx.x * blockDim.x + threadIdx.x;
    if (v >= vg::NVOX) return;

    int t = (int)v;
    const int z = t % vg::V; t /= vg::V;
    const int y = t % vg::V; t /= vg::V;
    const int x = t % vg::V;               // t/V == batch, unused

    float* p = out + (size_t)v * vg::C;
    // 40-byte records: 8-byte aligned always -> use b64 accesses.
    float2* p2 = reinterpret_cast<float2*>(p);
    const float2 s01 = p2[0];
    const float2 s23 = p2[1];
    const float2 s45 = p2[2];
    const float2 s89 = p2[4];              // .y = count

    const float cnt = s89.y;
    const float dn  = fmaxf(cnt, 1.0f);

    float2 o;
    o.x = s01.x / dn; o.y = s01.y / dn; p2[0] = o;   // mean coords xy
    o.x = s23.x / dn; o.y = s23.y / dn; p2[1] = o;   // mean coord z, feat0
    o.x = s45.x / dn; o.y = s45.y / dn; p2[2] = o;   // feat1, feat2
    o.x = (float)x / (float)vg::V;                   // index grid (meshgrid 'ij')
    o.y = (float)y / (float)vg::V;
    p2[3] = o;
    o.x = (float)z / (float)vg::V;
    o.y = (cnt > 0.0f) ? 1.0f : 0.0f;                // occupancy
    p2[4] = o;
}

extern "C" void kernel_launch(void* const* d_in, const int* in_sizes, int n_in,
                              void* d_out, int out_size, void* d_ws, size_t ws_size,
                              hipStream_t stream)
{
    (void)in_sizes; (void)n_in; (void)d_ws; (void)ws_size;
    const float* coords = (const float*)d_in[0];
    const float* feats  = (const float*)d_in[1];
    float* out = (float*)d_out;

    // (1) zero accumulators (also resets state for every graph replay)
    hipMemsetAsync(out, 0, (size_t)out_size * sizeof(float), stream);

    // (2) scatter-add points
    {
        const int threads = 256;
        const int blocks  = (int)((vg::NPTS + threads - 1) / threads);
        voxel_scatter<<<blocks, threads, 0, stream>>>(coords, feats, out);
    }

    // (3) finalize: mean + index grid + occupancy, in place
    {
        const int threads = 256;
        const int blocks  = (int)((vg::NVOX + threads - 1) / threads);
        voxel_finalize<<<blocks, threads, 0, stream>>>(out);
    }
}